// _MeanPoolBackbone_10608569221147
// MI455X (gfx1250) — compile-verified
//
#include <hip/hip_runtime.h>
#include <stdint.h>

// ---------------------------------------------------------------------------
// Segment-mean (sorted ids), MI455X / gfx1250.
// Bandwidth-bound (0.25 FLOP/byte): stream x once via CDNA5 async
// global->LDS DMA (ASYNCcnt path), double-buffered, accumulate in fp32 VALU.
// ---------------------------------------------------------------------------

__device__ __forceinline__ void async_ld_b128(uint32_t lds_byte_off, const void* gaddr) {
    // copies gaddr to LDS byte offset lds_byte_off, 16 bytes per lane, tracked by ASYNCcnt.
    asm volatile("global_load_async_to_lds_b128 %0, %1, off th:TH_LOAD_NT"
                 :: "v"(lds_byte_off), "v"(gaddr)
                 : "memory");
}

__device__ __forceinline__ void wait_async_le4() {
    asm volatile("s_wait_asynccnt 0x4" ::: "memory");
}
__device__ __forceinline__ void wait_async_le0() {
    asm volatile("s_wait_asynccnt 0x0" ::: "memory");
}

// Fast path: D == 256. One block (256 threads, 8 wave32) per segment.
__global__ __launch_bounds__(256) void segment_mean_async256(
    const float* __restrict__ x,
    const long long* __restrict__ ids,
    float* __restrict__ out,
    int N)
{
    constexpr int D = 256;
    constexpr int R = 16;                  // rows per chunk
    constexpr int CHUNK_FLOATS = R * D;    // 4096 floats = 16 KB

    __shared__ float sbuf[2][CHUNK_FLOATS];
    __shared__ int   sbounds[2];

    const int g = blockIdx.x;
    const int t = threadIdx.x;

    // --- segment bounds: lower_bound(g), lower_bound(g+1) on sorted ids ---
    if (t < 2) {
        const long long key = (long long)g + (long long)t;
        int lo = 0, hi = N;
        while (lo < hi) {
            int mid = (lo + hi) >> 1;
            if (ids[mid] < key) lo = mid + 1; else hi = mid;
        }
        sbounds[t] = lo;
    }
    __syncthreads();
    const int start = sbounds[0];
    const int end   = sbounds[1];
    const int count = end - start;

    const int nchunks = (count + R - 1) / R;

    // Issue one 16-row chunk: 4 x b128 per thread, uniform EXEC per wave.
    // Tail rows clamp the *source* row to the last valid one (dest slots for
    // clamped rows are never read), so ASYNCcnt bookkeeping stays uniform.
    auto issue_chunk = [&](int ci, int buf) {
        const int row0 = start + ci * R;
        const int rows = min(R, end - row0);                 // >= 1 by construction
        const float* gbase = x + (size_t)row0 * D;
        const uint32_t lds_base = (uint32_t)(uintptr_t)(&sbuf[buf][0]);
        #pragma unroll
        for (int j = 0; j < 4; ++j) {
            const int fidx = j * 1024 + 4 * t;               // float idx in chunk
            const int r    = fidx >> 8;                      // row in chunk
            const int rs   = (r < rows) ? r : (rows - 1);    // clamp source row
            const int sfx  = rs * D + (fidx & (D - 1));
            async_ld_b128(lds_base + (uint32_t)(fidx * 4), (const void*)(gbase + sfx));
        }
    };

    float acc = 0.0f;
    if (nchunks > 0) issue_chunk(0, 0);

    for (int ci = 0; ci < nchunks; ++ci) {
        if (ci + 1 < nchunks) {
            issue_chunk(ci + 1, (ci + 1) & 1);
            wait_async_le4();   // in-order: the 4 oldest (chunk ci) retired
        } else {
            wait_async_le0();
        }
        __syncthreads();        // make chunk ci visible across waves

        const float* lb  = &sbuf[ci & 1][0];
        const int rows   = min(R, end - (start + ci * R));
        #pragma unroll
        for (int r = 0; r < R; ++r) {
            if (r < rows) acc += lb[r * D + t];              // stride-256: bank-conflict-free
        }
        __syncthreads();        // done reading before this buffer is re-filled
    }

    out[(size_t)g * D + t] = acc / (float)count;             // 0/0 -> NaN matches reference
}

// Generic fallback for D != 256 (insurance only; reference uses D = 256).
__global__ __launch_bounds__(256) void segment_mean_generic(
    const float* __restrict__ x,
    const long long* __restrict__ ids,
    float* __restrict__ out,
    int N, int D)
{
    __shared__ int sbounds[2];
    const int g = blockIdx.x;
    if (threadIdx.x < 2) {
        const long long key = (long long)g + (long long)threadIdx.x;
        int lo = 0, hi = N;
        while (lo < hi) {
            int mid = (lo + hi) >> 1;
            if (ids[mid] < key) lo = mid + 1; else hi = mid;
        }
        sbounds[threadIdx.x] = lo;
    }
    __syncthreads();
    const int start = sbounds[0];
    const int end   = sbounds[1];
    const float cnt = (float)(end - start);

    for (int c = threadIdx.x; c < D; c += blockDim.x) {
        float acc = 0.0f;
        for (int r = start; r < end; ++r) acc += x[(size_t)r * D + c];
        out[(size_t)g * D + c] = acc / cnt;
    }
}

extern "C" void kernel_launch(void* const* d_in, const int* in_sizes, int n_in,
                              void* d_out, int out_size, void* d_ws, size_t ws_size,
                              hipStream_t stream) {
    const float*     x   = (const float*)d_in[0];
    const long long* ids = (const long long*)d_in[1];   // int64 batch_ids
    float*           out = (float*)d_out;

    const int N = in_sizes[1];
    const int D = (N > 0) ? (in_sizes[0] / N) : 0;
    const int G = (D > 0) ? (out_size / D) : 0;
    if (G <= 0) return;

    if (D == 256) {
        segment_mean_async256<<<dim3(G), dim3(256), 0, stream>>>(x, ids, out, N);
    } else {
        segment_mean_generic<<<dim3(G), dim3(256), 0, stream>>>(x, ids, out, N, D);
    }
}